// GNNModule_51075751084149
// MI455X (gfx1250) — compile-verified
//
#include <hip/hip_runtime.h>

// GNN layer for MI455X (gfx1250, wave32, WMMA).
//
// Roofline: streaming traffic is ~200 MB (x/adj/out all L2-resident at 192 MB
// L2; ~10 us at 23.3 TB/s), so the only thing that can dominate is atomic RMW
// traffic. The naive segment-sum SpMM needs 204.8M fp32 L2 atomics; instead we
// build CSR on-device (~3.2M int atomics total) and run both SpMMs as pure
// gather-sum kernels (no atomics, every output written exactly once -> no
// clearing of the 51 MB adj buffers either).
//
// Pipeline (all on `stream`):
//   1) zero_deg          : clear degree histogram (N ints)
//   2) deg_kernel        : deg[row[e]]++                  (E int atomics)
//   3) scan_block/top/add: exclusive scan deg -> rowptr, cursor=rowptr
//   4) scatter_kernel    : CSR (col_s, val_s) via cursor  (E int atomics)
//   5) pack_kernel       : fold Wa0+1+2 / Wb0+1+2 and repack all 6 weight
//                          matrices into per-lane WMMA B-fragment layout
//   6) spmm_csr x2       : adj0 = A@x ; adj1 = A@adj0  (gather, float2/lane)
//   7) gemm_bn_kernel    : per 16-row block, 384 v_wmma_f32_16x16x4_f32
//                          + fused ReLU/BN epilogue, writes [N,128] output.

#define N_NODES 100000
#define N_EDGES 1600000
#define D_FEAT  64
#define BN_EPS  1e-3f

typedef float v2f __attribute__((ext_vector_type(2)));
typedef float v8f __attribute__((ext_vector_type(8)));

// ---------------------------------------------------------------------------
__global__ void zero_deg_kernel(int* __restrict__ deg) {
    unsigned i = blockIdx.x * 256u + threadIdx.x;
    if (i < N_NODES) deg[i] = 0;
}

// deg[row[e]]++  (grid sized exactly E/256)
__global__ void deg_kernel(const int* __restrict__ row, int* __restrict__ deg) {
    unsigned e = blockIdx.x * 256u + threadIdx.x;
    atomicAdd(&deg[row[e]], 1);
}

// ---------------------------------------------------------------------------
// Exclusive scan of deg[N] -> rowptr[N], 1024 elements per block (256 thr x 4),
// block totals into partial[]. 98 blocks cover N=100000.
__global__ void scan_block_kernel(const int* __restrict__ deg,
                                  int* __restrict__ rowptr,
                                  int* __restrict__ partial) {
    __shared__ int lsum[256];
    const unsigned t    = threadIdx.x;
    const unsigned base = blockIdx.x * 1024u + t * 4u;
    int d0 = (base + 0 < N_NODES) ? deg[base + 0] : 0;
    int d1 = (base + 1 < N_NODES) ? deg[base + 1] : 0;
    int d2 = (base + 2 < N_NODES) ? deg[base + 2] : 0;
    int d3 = (base + 3 < N_NODES) ? deg[base + 3] : 0;
    lsum[t] = d0 + d1 + d2 + d3;
    __syncthreads();
    // Hillis-Steele inclusive scan over 256 partial sums
    for (unsigned off = 1; off < 256; off <<= 1) {
        int v = (t >= off) ? lsum[t - off] : 0;
        __syncthreads();
        lsum[t] += v;
        __syncthreads();
    }
    if (t == 255) partial[blockIdx.x] = lsum[255];
    int excl = (t == 0) ? 0 : lsum[t - 1];
    if (base + 0 < N_NODES) rowptr[base + 0] = excl;
    if (base + 1 < N_NODES) rowptr[base + 1] = excl + d0;
    if (base + 2 < N_NODES) rowptr[base + 2] = excl + d0 + d1;
    if (base + 3 < N_NODES) rowptr[base + 3] = excl + d0 + d1 + d2;
}

// Serial exclusive scan of the (<=128) block totals: tiny, one thread.
__global__ void scan_top_kernel(int* __restrict__ partial, int nblocks) {
    if (blockIdx.x == 0 && threadIdx.x == 0) {
        int run = 0;
        for (int i = 0; i < nblocks; ++i) {
            int v = partial[i];
            partial[i] = run;
            run += v;
        }
    }
}

// Add block offsets; initialize scatter cursors; close rowptr.
__global__ void scan_add_kernel(int* __restrict__ rowptr,
                                int* __restrict__ cursor,
                                const int* __restrict__ partial) {
    unsigned i = blockIdx.x * 256u + threadIdx.x;
    if (i < N_NODES) {
        int v = rowptr[i] + partial[i >> 10];
        rowptr[i] = v;
        cursor[i] = v;
    }
    if (i == 0) rowptr[N_NODES] = N_EDGES;
}

// CSR scatter: place each edge's (col, val) into its row's segment.
__global__ void scatter_kernel(const int* __restrict__ row,
                               const int* __restrict__ col,
                               const float* __restrict__ vals,
                               int* __restrict__ cursor,
                               int* __restrict__ col_s,
                               float* __restrict__ val_s) {
    unsigned e = blockIdx.x * 256u + threadIdx.x;   // grid sized exactly E/256
    int r = row[e];
    int pos = atomicAdd(&cursor[r], 1);
    col_s[pos] = col[e];
    val_s[pos] = vals[e];
}

// ---------------------------------------------------------------------------
// Gather SpMM: one wave per node, each lane owns 2 features (float2).
// y[n,:] = sum_{e in rowptr[n]..rowptr[n+1]} val_s[e] * x[col_s[e],:]
__launch_bounds__(256)
__global__ void spmm_csr_kernel(const int* __restrict__ rowptr,
                                const int* __restrict__ col_s,
                                const float* __restrict__ val_s,
                                const float* __restrict__ xin,
                                float* __restrict__ yout) {
    const unsigned node = (blockIdx.x * 256u + threadIdx.x) >> 5;  // 12500*8 = 100000 exact
    const unsigned lane = threadIdx.x & 31u;
    const int s = rowptr[node];
    const int t = rowptr[node + 1];
    v2f acc = {0.f, 0.f};
    for (int e = s; e < t; ++e) {
        int   c = col_s[e];
        float v = val_s[e];
        v2f xv = *(const v2f*)(xin + (size_t)c * 64 + lane * 2u);
        acc.x = fmaf(v, xv.x, acc.x);
        acc.y = fmaf(v, xv.y, acc.y);
    }
    *(v2f*)(yout + (size_t)node * 64 + lane * 2u) = acc;
}

// ---------------------------------------------------------------------------
// Repack weights into WMMA B-fragment layout.
// Fragment layout (mirror of documented 16x4 A layout):
//   lane L: n = L&15, half = L>>4; b.x = W[k*4+2*half][col], b.y = W[k*4+2*half+1][col]
// pack[mat][k][j][lane] as float2, mat in {Wa0+1+2, Wa3, Wa4, Wb0+1+2, Wb3, Wb4}
__global__ void pack_kernel(const float* __restrict__ Wa,
                            const float* __restrict__ Wb,
                            float2* __restrict__ pack) {
    unsigned tid  = blockIdx.x * 256u + threadIdx.x;   // 12288 total
    unsigned mat  = tid >> 11;
    unsigned rem  = tid & 2047u;
    unsigned k    = rem >> 7;
    unsigned rem2 = rem & 127u;
    unsigned j    = rem2 >> 5;
    unsigned lane = rem2 & 31u;
    unsigned half = lane >> 4;
    unsigned n    = lane & 15u;
    unsigned row0 = k * 4u + 2u * half;
    unsigned col  = j * 16u + n;

    const float* W = (mat < 3u) ? Wa : Wb;
    unsigned m = mat % 3u;
    float2 o;
    if (m == 0u) {
        o.x = W[0 * 4096 + row0 * 64 + col] + W[1 * 4096 + row0 * 64 + col] +
              W[2 * 4096 + row0 * 64 + col];
        o.y = W[0 * 4096 + (row0 + 1) * 64 + col] + W[1 * 4096 + (row0 + 1) * 64 + col] +
              W[2 * 4096 + (row0 + 1) * 64 + col];
    } else {
        unsigned s = m + 2u;
        o.x = W[s * 4096 + row0 * 64 + col];
        o.y = W[s * 4096 + (row0 + 1) * 64 + col];
    }
    pack[tid] = o;
}

// ---------------------------------------------------------------------------
// One wave per 16-row block: 8 output 16x16 tiles (4 alpha + 4 beta),
// K=64 -> 16 k-steps of V_WMMA_F32_16X16X4_F32, 24 WMMAs per k-step.
__launch_bounds__(64)
__global__ void gemm_bn_kernel(const float* __restrict__ x,
                               const float* __restrict__ adj0,
                               const float* __restrict__ adj1,
                               const float2* __restrict__ pack,
                               const float* __restrict__ gamma_a,
                               const float* __restrict__ beta_a,
                               const float* __restrict__ mean_a,
                               const float* __restrict__ var_a,
                               const float* __restrict__ gamma_b,
                               const float* __restrict__ beta_b,
                               const float* __restrict__ mean_b,
                               const float* __restrict__ var_b,
                               float* __restrict__ out) {
    const unsigned lane = threadIdx.x & 31u;
    const unsigned wave = threadIdx.x >> 5;
    const unsigned half = lane >> 4;
    const unsigned n    = lane & 15u;
    const unsigned r0   = (blockIdx.x * 2u + wave) * 16u;   // 100000/16 = 6250 waves exact

    // Fused BN constants per column tile: out = h*scale + shift
    float sa[4], ba[4], sb[4], bb[4];
#pragma unroll
    for (int j = 0; j < 4; ++j) {
        unsigned ch = (unsigned)j * 16u + n;
        float s = gamma_a[ch] * rsqrtf(var_a[ch] + BN_EPS);
        sa[j] = s;
        ba[j] = beta_a[ch] - mean_a[ch] * s;
        float t = gamma_b[ch] * rsqrtf(var_b[ch] + BN_EPS);
        sb[j] = t;
        bb[j] = beta_b[ch] - mean_b[ch] * t;
    }

    v8f acc_a[4] = {};
    v8f acc_b[4] = {};

    const unsigned rowm = r0 + n;
    const float* xr  = x    + (size_t)rowm * 64;
    const float* a0r = adj0 + (size_t)rowm * 64;
    const float* a1r = adj1 + (size_t)rowm * 64;

#pragma unroll 4
    for (unsigned k = 0; k < 16; ++k) {
        const unsigned koff = k * 4u + 2u * half;
        v2f ax = *(const v2f*)(xr  + koff);
        v2f a0 = *(const v2f*)(a0r + koff);
        v2f a1 = *(const v2f*)(a1r + koff);
#pragma unroll
        for (int j = 0; j < 4; ++j) {
            const unsigned base = (k * 4u + (unsigned)j) * 32u + lane;
            v2f bwsa = *(const v2f*)&pack[0 * 2048 + base];
            v2f bwa3 = *(const v2f*)&pack[1 * 2048 + base];
            v2f bwa4 = *(const v2f*)&pack[2 * 2048 + base];
            v2f bwsb = *(const v2f*)&pack[3 * 2048 + base];
            v2f bwb3 = *(const v2f*)&pack[4 * 2048 + base];
            v2f bwb4 = *(const v2f*)&pack[5 * 2048 + base];
            acc_a[j] = __builtin_amdgcn_wmma_f32_16x16x4_f32(false, ax, false, bwsa,
                                                             (short)0, acc_a[j], false, false);
            acc_a[j] = __builtin_amdgcn_wmma_f32_16x16x4_f32(false, a0, false, bwa3,
                                                             (short)0, acc_a[j], false, false);
            acc_a[j] = __builtin_amdgcn_wmma_f32_16x16x4_f32(false, a1, false, bwa4,
                                                             (short)0, acc_a[j], false, false);
            acc_b[j] = __builtin_amdgcn_wmma_f32_16x16x4_f32(false, ax, false, bwsb,
                                                             (short)0, acc_b[j], false, false);
            acc_b[j] = __builtin_amdgcn_wmma_f32_16x16x4_f32(false, a0, false, bwb3,
                                                             (short)0, acc_b[j], false, false);
            acc_b[j] = __builtin_amdgcn_wmma_f32_16x16x4_f32(false, a1, false, bwb4,
                                                             (short)0, acc_b[j], false, false);
        }
    }

    // Epilogue: C/D tile layout: VGPR v, lane L -> row = v + 8*(L>>4), col = L&15
#pragma unroll
    for (int j = 0; j < 4; ++j) {
#pragma unroll
        for (int v = 0; v < 8; ++v) {
            size_t row = (size_t)r0 + (unsigned)v + 8u * half;
            float va = acc_a[j][v];
            va = va > 0.f ? va : 0.f;   // ReLU before BN (alpha path)
            out[row * 128 + (size_t)j * 16 + n]      = fmaf(va, sa[j], ba[j]);
            out[row * 128 + 64 + (size_t)j * 16 + n] = fmaf(acc_b[j][v], sb[j], bb[j]);
        }
    }
}

// ---------------------------------------------------------------------------
extern "C" void kernel_launch(void* const* d_in, const int* in_sizes, int n_in,
                              void* d_out, int out_size, void* d_ws, size_t ws_size,
                              hipStream_t stream) {
    const float* x       = (const float*)d_in[0];
    const int*   erow    = (const int*)d_in[1];
    const int*   ecol    = (const int*)d_in[2];
    const float* evals   = (const float*)d_in[3];
    const float* Wa      = (const float*)d_in[4];   // [5,64,64]
    const float* Wb      = (const float*)d_in[5];   // [5,64,64]
    const float* gamma_a = (const float*)d_in[6];
    const float* beta_a  = (const float*)d_in[7];
    const float* mean_a  = (const float*)d_in[8];
    const float* var_a   = (const float*)d_in[9];
    const float* gamma_b = (const float*)d_in[10];
    const float* beta_b  = (const float*)d_in[11];
    const float* mean_b  = (const float*)d_in[12];
    const float* var_b   = (const float*)d_in[13];
    float* out = (float*)d_out;

    // Workspace layout:
    //   adj0 [N*64 f32] | adj1 [N*64 f32] | pack [6*4096 f32]
    //   rowptr [N+1 i32] | cursor/deg [N i32] | partial [128 i32]
    //   col_s [E i32] | val_s [E f32]                     (~64.9 MB total)
    float*  adj0    = (float*)d_ws;
    float*  adj1    = adj0 + (size_t)N_NODES * D_FEAT;
    float*  packf   = adj1 + (size_t)N_NODES * D_FEAT;
    int*    rowptr  = (int*)(packf + 6 * 4096);
    int*    cursor  = rowptr + (N_NODES + 1);      // doubles as deg histogram
    int*    partial = cursor + N_NODES;
    int*    col_s   = partial + 128;
    float*  val_s   = (float*)(col_s + N_EDGES);
    float2* pack    = (float2*)packf;

    const unsigned nblk_N  = (N_NODES + 255) / 256;     // 391
    const unsigned nblk_E  = N_EDGES / 256;             // 6250 exact
    const int      nscan   = (N_NODES + 1023) / 1024;   // 98

    // --- CSR build (replaces 204.8M fp32 atomics with ~3.2M int atomics) ---
    zero_deg_kernel<<<nblk_N, 256, 0, stream>>>(cursor);
    deg_kernel<<<nblk_E, 256, 0, stream>>>(erow, cursor);
    scan_block_kernel<<<nscan, 256, 0, stream>>>(cursor, rowptr, partial);
    scan_top_kernel<<<1, 32, 0, stream>>>(partial, nscan);
    scan_add_kernel<<<nblk_N, 256, 0, stream>>>(rowptr, cursor, partial);
    scatter_kernel<<<nblk_E, 256, 0, stream>>>(erow, ecol, evals, cursor, col_s, val_s);

    // --- weight fold + WMMA B-fragment repack ---
    pack_kernel<<<48, 256, 0, stream>>>(Wa, Wb, pack);

    // --- gather SpMMs: adj0 = A@x ; adj1 = A@adj0 (no atomics, no clears) ---
    spmm_csr_kernel<<<N_NODES / 8, 256, 0, stream>>>(rowptr, col_s, val_s, x, adj0);
    spmm_csr_kernel<<<N_NODES / 8, 256, 0, stream>>>(rowptr, col_s, val_s, adj0, adj1);

    // --- fused WMMA GEMM + BN epilogue: 2 waves/block, 16 rows/wave ---
    gemm_bn_kernel<<<N_NODES / 32, 64, 0, stream>>>(x, adj0, adj1, pack,
                                                    gamma_a, beta_a, mean_a, var_a,
                                                    gamma_b, beta_b, mean_b, var_b,
                                                    out);
}